// Attn_30700426232094
// MI455X (gfx1250) — compile-verified
//
#include <hip/hip_runtime.h>

// Problem sizes (fixed by the reference)
#define S_TOK 32768
#define HDIM  1024

typedef __attribute__((ext_vector_type(2))) float v2f;  // A/B frag: 16x4 f32 = 2 VGPRs
typedef __attribute__((ext_vector_type(8))) float v8f;  // C/D frag: 16x16 f32 = 8 VGPRs

// d_ws float layout
#define WS_U    0       // u[1024] = W^T v
#define WS_C    1024    // c = v.b
#define WS_PMAX 1056    // per-block maxima [256]
#define WS_PSUM 1312    // per-block exp-sums [128]

// ---------------------------------------------------------------------------
// K1: u[h] = sum_o v[o] * W[o*H + h]  (W row-major [out,in]);  c = v.b
// 32 blocks x 256 threads; h = blk*32 + tid%32 (coalesced), 8-way o split.
// ---------------------------------------------------------------------------
__global__ __launch_bounds__(256) void k_prep(const float* __restrict__ W,
                                              const float* __restrict__ b,
                                              const float* __restrict__ v,
                                              float* __restrict__ ws) {
  __shared__ float s[256];
  const int tid = threadIdx.x;
  const int h = blockIdx.x * 32 + (tid & 31);
  const int obase = (tid >> 5) * 128;
  float acc = 0.f;
#pragma unroll 4
  for (int j = 0; j < 128; ++j) {
    const int o = obase + j;
    acc = fmaf(v[o], W[o * HDIM + h], acc);
  }
  s[tid] = acc;
  __syncthreads();
  if (tid < 128) s[tid] += s[tid + 128];
  __syncthreads();
  if (tid < 64) s[tid] += s[tid + 64];
  __syncthreads();
  if (tid < 32) ws[WS_U + blockIdx.x * 32 + tid] = s[tid] + s[tid + 32];

  if (blockIdx.x == 0) {  // block-uniform branch: safe to sync inside
    __syncthreads();
    float p = 0.f;
    for (int j = tid; j < HDIM; j += 256) p = fmaf(v[j], b[j], p);
    s[tid] = p;
    __syncthreads();
    for (int st = 128; st > 0; st >>= 1) {
      if (tid < st) s[tid] += s[tid + st];
      __syncthreads();
    }
    if (tid == 0) ws[WS_C] = s[0];
  }
}

// ---------------------------------------------------------------------------
// K2: e[s] = Q[s,:].u + c via V_WMMA_F32_16X16X4_F32, 16 rows per wave.
// A frag (16x4 f32): lane L<16 -> row L, K = k+{0,1}; lane L>=16 -> row L-16,
// K = k+{2,3}. B frag: same K pattern of u, replicated over all 16 columns,
// so C[m,n] = e[r0+m] for every n.
//
// Software-pipelined: 16 bodies of K=64 (16 WMMAs each). Body i's 32 loads
// land in distinct registers (buffer i&1) while body i-1 computes, so the
// scheduler can issue partial s_wait_loadcnt instead of draining to zero per
// WMMA (the Round-1 bottleneck: one register pair reused for every load).
// ---------------------------------------------------------------------------
__global__ __launch_bounds__(256) void k_energy(const float* __restrict__ q,
                                                const float* __restrict__ ws,
                                                float* __restrict__ eout,
                                                float* __restrict__ pmax) {
  __shared__ float smax[8];
  const int tid = threadIdx.x;
  const int lane = tid & 31;
  const int wv = tid >> 5;
  const int r0 = (blockIdx.x * 8 + wv) * 16;
  const int mrow = lane & 15;
  const int khalf = lane >> 4;            // 0: K+{0,1}, 1: K+{2,3}
  const int koff = khalf * 2;
  const float cval = ws[WS_C];
  const float* __restrict__ qp = q + (size_t)(r0 + mrow) * HDIM + koff;
  const float* __restrict__ up = ws + WS_U + koff;

  // Double-buffered fragment staging; all indices compile-time after unroll,
  // so these arrays live entirely in VGPRs.
  v2f qa[2][16];
  v2f ub[2][16];

#pragma unroll
  for (int j = 0; j < 16; ++j) {
    qa[0][j] = *(const v2f*)(qp + j * 4);
    ub[0][j] = *(const v2f*)(up + j * 4);
  }

  v8f acc0 = {};
  v8f acc1 = {};

#pragma unroll
  for (int blk = 0; blk < 16; ++blk) {
    const int cur = blk & 1;
    const int nxt = cur ^ 1;
    if (blk < 15) {                       // constant-folds under full unroll
      const int kb = (blk + 1) * 64;
#pragma unroll
      for (int j = 0; j < 16; ++j) {
        qa[nxt][j] = *(const v2f*)(qp + kb + j * 4);
        ub[nxt][j] = *(const v2f*)(up + kb + j * 4);
      }
    }
#pragma unroll
    for (int j = 0; j < 16; j += 2) {
      acc0 = __builtin_amdgcn_wmma_f32_16x16x4_f32(false, qa[cur][j], false,
                                                   ub[cur][j], (short)0, acc0,
                                                   false, false);
      acc1 = __builtin_amdgcn_wmma_f32_16x16x4_f32(false, qa[cur][j + 1], false,
                                                   ub[cur][j + 1], (short)0,
                                                   acc1, false, false);
    }
  }

  const float e0 = acc0[0] + acc1[0] + cval;
  const float e1 = acc0[1] + acc1[1] + cval;
  const float e2 = acc0[2] + acc1[2] + cval;
  const float e3 = acc0[3] + acc1[3] + cval;
  const float e4 = acc0[4] + acc1[4] + cval;
  const float e5 = acc0[5] + acc1[5] + cval;
  const float e6 = acc0[6] + acc1[6] + cval;
  const float e7 = acc0[7] + acc1[7] + cval;

  // Lane 0 holds rows r0..r0+7, lane 16 holds rows r0+8..r0+15 (any column).
  if (mrow == 0) {
    float* dst = eout + r0 + khalf * 8;
    float4 lo; lo.x = e0; lo.y = e1; lo.z = e2; lo.w = e3;
    float4 hi; hi.x = e4; hi.y = e5; hi.z = e6; hi.w = e7;
    *(float4*)(dst) = lo;
    *(float4*)(dst + 4) = hi;
  }

  // Per-block max for softmax.
  float m = fmaxf(fmaxf(fmaxf(e0, e1), fmaxf(e2, e3)),
                  fmaxf(fmaxf(e4, e5), fmaxf(e6, e7)));
#pragma unroll
  for (int off = 16; off > 0; off >>= 1) m = fmaxf(m, __shfl_xor(m, off, 32));
  if (lane == 0) smax[wv] = m;
  __syncthreads();
  if (tid == 0) {
    float bm = smax[0];
#pragma unroll
    for (int i = 1; i < 8; ++i) bm = fmaxf(bm, smax[i]);
    pmax[blockIdx.x] = bm;
  }
}

// ---------------------------------------------------------------------------
// K3: every block redundantly reduces the 256 partial maxima -> gmax,
// computes exp(e - gmax) in place, and writes its deterministic block sum.
// ---------------------------------------------------------------------------
__global__ __launch_bounds__(256) void k_exp(float* __restrict__ e,
                                             const float* __restrict__ pmax,
                                             float* __restrict__ psum) {
  __shared__ float s[256];
  const int tid = threadIdx.x;
  s[tid] = pmax[tid];
  __syncthreads();
  for (int st = 128; st > 0; st >>= 1) {
    if (tid < st) s[tid] = fmaxf(s[tid], s[tid + st]);
    __syncthreads();
  }
  const float gmax = s[0];
  __syncthreads();

  const int i = blockIdx.x * 256 + tid;
  const float x = expf(e[i] - gmax);
  e[i] = x;
  s[tid] = x;
  __syncthreads();
  for (int st = 128; st > 0; st >>= 1) {
    if (tid < st) s[tid] += s[tid + st];
    __syncthreads();
  }
  if (tid == 0) psum[blockIdx.x] = s[0];
}

// ---------------------------------------------------------------------------
// K4: every block redundantly reduces the 128 partial sums and normalizes.
// ---------------------------------------------------------------------------
__global__ __launch_bounds__(256) void k_norm(float* __restrict__ out,
                                              const float* __restrict__ psum) {
  __shared__ float s[128];
  const int tid = threadIdx.x;
  if (tid < 128) s[tid] = psum[tid];
  __syncthreads();
  for (int st = 64; st > 0; st >>= 1) {
    if (tid < st) s[tid] += s[tid + st];
    __syncthreads();
  }
  const float inv = 1.0f / s[0];
  const int i = blockIdx.x * 256 + tid;
  out[i] = out[i] * inv;
}

// ---------------------------------------------------------------------------
extern "C" void kernel_launch(void* const* d_in, const int* in_sizes, int n_in,
                              void* d_out, int out_size, void* d_ws, size_t ws_size,
                              hipStream_t stream) {
  const float* q = (const float*)d_in[0];  // question [32768,1024]
  const float* W = (const float*)d_in[1];  // W [1024,1024] row-major [out,in]
  const float* b = (const float*)d_in[2];  // b [1024]
  const float* v = (const float*)d_in[3];  // v [1024]
  float* out = (float*)d_out;              // [32768] softmax, staged as energies
  float* ws = (float*)d_ws;

  k_prep<<<32, 256, 0, stream>>>(W, b, v, ws);
  k_energy<<<256, 256, 0, stream>>>(q, ws, out, ws + WS_PMAX);
  k_exp<<<128, 256, 0, stream>>>(out, ws + WS_PMAX, ws + WS_PSUM);
  k_norm<<<128, 256, 0, stream>>>(out, ws + WS_PSUM);
}